// PhysicsGatedAttention_40492951666854
// MI455X (gfx1250) — compile-verified
//
#include <hip/hip_runtime.h>
#include <math.h>

// Problem constants (PhysicsGatedAttention): B=2, N=512, HIDDEN=256, HEADS=4, HEAD_DIM=64
#define N_   512
#define D_   256
#define H_   4
#define HD_  64
#define B_   2
#define SCALE   0.125f      // HEAD_DIM^-0.5
#define NEGINF  (-1e9f)

typedef float v2f __attribute__((ext_vector_type(2)));
typedef float v8f __attribute__((ext_vector_type(8)));

// V_WMMA_F32_16X16X4_F32 : D(16x16 f32) = A(16x4 f32) * B(4x16 f32) + C
// Layouts (wave32):
//   A frag: lane = m + 16*g (m=0..15, g=0..1); a.x=A[m][k+2g], a.y=A[m][k+2g+1]
//   B frag: lane = n + 16*g;                    b.x=B[k+2g][n], b.y=B[k+2g+1][n]
//   C/D   : lane = n + 16*g, vgpr p => D[p+8g][n]
__device__ __forceinline__ v8f wmma4(v2f a, v2f b, v8f c) {
  return __builtin_amdgcn_wmma_f32_16x16x4_f32(false, a, false, b, (short)0, c,
                                               false, false);
}

__device__ __forceinline__ float gelu_exact(float v) {
  // jax.nn.gelu(approximate=False): 0.5*x*(1+erf(x/sqrt(2)))
  return 0.5f * v * (1.0f + erff(v * 0.70710678118654752440f));
}

// ---------------------------------------------------------------------------
// Generic fp32 GEMM: Y[M,Nc] = X[M,K] @ W[K,Nc] (+ bias[Nc]).
// One wave per 16x16 output tile, 8 waves per block. M,K,Nc multiples of 16/4.
// ---------------------------------------------------------------------------
__global__ __launch_bounds__(256) void gemm_xw_kernel(
    const float* __restrict__ X, const float* __restrict__ W,
    const float* __restrict__ bias, float* __restrict__ Y,
    int M, int K, int Nc) {
  const int wave = threadIdx.x >> 5;
  const int lane = threadIdx.x & 31;
  const int m16 = lane & 15;
  const int g   = lane >> 4;
  const int ntiles = Nc >> 4;
  const int tile = blockIdx.x * 8 + wave;
  const int mt = tile / ntiles;
  const int nt = tile - mt * ntiles;

  const float* xrow = X + (size_t)(mt * 16 + m16) * K;   // A row m
  const float* wcol = W + (nt * 16 + m16);               // B column n
  v8f acc = {};
  for (int kk = 0; kk < K; kk += 4) {
    const int d0 = kk + 2 * g;
    v2f a = *(const v2f*)(xrow + d0);                    // contiguous pair
    v2f b;
    b.x = wcol[(size_t)d0 * Nc];
    b.y = wcol[(size_t)(d0 + 1) * Nc];
    acc = wmma4(a, b, acc);
  }
  const int col = nt * 16 + m16;
  const float bv = bias ? bias[col] : 0.0f;
  float* yout = Y + (size_t)(mt * 16) * Nc + col;
#pragma unroll
  for (int p = 0; p < 8; ++p)
    yout[(size_t)(p + 8 * g) * Nc] = acc[p] + bv;
}

// ---------------------------------------------------------------------------
// Phase A: S[b,h,i,j] = scale*(q_i . k_j) + (mask ? physics_bias : -1e9)
// One block per (b, i-tile16, j-tile16). 8 waves:
//   waves 0..3 : QK^T for head = wave (16 WMMA over hd=64)
//   all waves  : physics bias for i = it*16 + 2*wave + {0,1}
//                (GELU into A-frag, [16j x 256d] @ Wp2[256d x 4h] via WMMA)
// ---------------------------------------------------------------------------
__global__ __launch_bounds__(256) void scores_kernel(
    const float* __restrict__ Q, const float* __restrict__ Kb,
    const float* __restrict__ Ab, const float* __restrict__ Bb,
    const float* __restrict__ Wp2, const float* __restrict__ bp2,
    const int* __restrict__ adj, float* __restrict__ S) {
  __shared__ float lds_qk[H_][16][16];     // [h][i][j]
  __shared__ float lds_bias[16][16][H_];   // [i][j][h]

  const int blk = blockIdx.x;              // b*1024 + it*32 + jt
  const int b  = blk >> 10;
  const int it = (blk >> 5) & 31;
  const int jt = blk & 31;
  const int wave = threadIdx.x >> 5;
  const int lane = threadIdx.x & 31;
  const int m16 = lane & 15;
  const int g   = lane >> 4;

  // ---- QK^T (heads on waves 0..3) ----
  if (wave < H_) {
    const int h = wave;
    const float* qrow = Q  + (size_t)(b * N_ + it * 16 + m16) * D_ + h * HD_;
    const float* krow = Kb + (size_t)(b * N_ + jt * 16 + m16) * D_ + h * HD_;
    v8f acc = {};
    for (int kk = 0; kk < HD_; kk += 4) {
      const int d0 = kk + 2 * g;
      v2f a = *(const v2f*)(qrow + d0);
      v2f bf = *(const v2f*)(krow + d0);   // B[k][n] = K[n][k] -> contiguous in k
      acc = wmma4(a, bf, acc);
    }
#pragma unroll
    for (int p = 0; p < 8; ++p) lds_qk[h][p + 8 * g][m16] = acc[p];
  }

  // ---- physics bias: every wave handles i-rows 2*wave and 2*wave+1 ----
  const float* brow = Bb + (size_t)(b * N_ + jt * 16 + m16) * D_;  // x_j side
#pragma unroll
  for (int s = 0; s < 2; ++s) {
    const int ii = wave * 2 + s;
    const float* arow = Ab + (size_t)(b * N_ + it * 16 + ii) * D_; // x_i side (+bp1)
    v8f acc = {};
    for (int kk = 0; kk < D_; kk += 4) {
      const int d0 = kk + 2 * g;
      v2f bp = *(const v2f*)(brow + d0);
      v2f a;
      a.x = gelu_exact(arow[d0]     + bp.x);
      a.y = gelu_exact(arow[d0 + 1] + bp.y);
      v2f bf;
      bf.x = (m16 < H_) ? Wp2[(size_t)d0 * H_ + m16]       : 0.0f;
      bf.y = (m16 < H_) ? Wp2[(size_t)(d0 + 1) * H_ + m16] : 0.0f;
      acc = wmma4(a, bf, acc);
    }
    if (m16 < H_) {
#pragma unroll
      for (int p = 0; p < 8; ++p) lds_bias[ii][p + 8 * g][m16] = acc[p];
    }
  }
  __syncthreads();

  // ---- combine + mask + store ----
  const int ii = threadIdx.x >> 4;
  const int jj = threadIdx.x & 15;
  const int i = it * 16 + ii;
  const int j = jt * 16 + jj;
  const bool mk = adj[i * N_ + j] != 0;
#pragma unroll
  for (int h = 0; h < H_; ++h) {
    const float bias_v = lds_bias[ii][jj][h] + bp2[h];
    const float v = lds_qk[h][ii][jj] * SCALE + (mk ? bias_v : NEGINF);
    S[((size_t)(b * H_ + h) * N_ + i) * N_ + j] = v;
  }
}

// ---------------------------------------------------------------------------
// Phase B: row softmax over j. One wave per row (512 elems -> 16/lane).
// ---------------------------------------------------------------------------
__global__ __launch_bounds__(256) void softmax_kernel(float* __restrict__ S) {
  const int wave = threadIdx.x >> 5;
  const int lane = threadIdx.x & 31;
  float* p = S + ((size_t)blockIdx.x * 8 + wave) * N_;

  float vals[16];
  float mx = -3.4e38f;
#pragma unroll
  for (int t = 0; t < 16; ++t) {
    vals[t] = p[lane + t * 32];
    mx = fmaxf(mx, vals[t]);
  }
#pragma unroll
  for (int off = 16; off >= 1; off >>= 1) mx = fmaxf(mx, __shfl_xor(mx, off));
  float sum = 0.0f;
#pragma unroll
  for (int t = 0; t < 16; ++t) {
    vals[t] = __expf(vals[t] - mx);
    sum += vals[t];
  }
#pragma unroll
  for (int off = 16; off >= 1; off >>= 1) sum += __shfl_xor(sum, off);
  const float inv = 1.0f / sum;
#pragma unroll
  for (int t = 0; t < 16; ++t) p[lane + t * 32] = vals[t] * inv;
}

// ---------------------------------------------------------------------------
// Phase C: O[b,i,h*64+d] = sum_j attn[b,h,i,j] * V[b,j,h*64+d]  (WMMA, K=512)
// One wave per 16x16 output tile: 8 (b,h) * 32 i-tiles * 4 d-tiles = 1024 tiles.
// ---------------------------------------------------------------------------
__global__ __launch_bounds__(256) void pv_kernel(
    const float* __restrict__ S, const float* __restrict__ V,
    float* __restrict__ O) {
  const int wave = threadIdx.x >> 5;
  const int lane = threadIdx.x & 31;
  const int m16 = lane & 15;
  const int g   = lane >> 4;
  const int tile = blockIdx.x * 8 + wave;
  const int nt = tile & 3;
  const int it = (tile >> 2) & 31;
  const int bh = tile >> 7;
  const int b = bh >> 2, h = bh & 3;

  const float* srow = S + ((size_t)bh * N_ + it * 16 + m16) * N_;
  const float* vcol = V + (size_t)b * N_ * D_ + h * HD_ + nt * 16 + m16;
  v8f acc = {};
  for (int kk = 0; kk < N_; kk += 4) {
    const int k0 = kk + 2 * g;
    v2f a = *(const v2f*)(srow + k0);
    v2f bf;
    bf.x = vcol[(size_t)k0 * D_];
    bf.y = vcol[(size_t)(k0 + 1) * D_];
    acc = wmma4(a, bf, acc);
  }
  float* orow = O + (size_t)(b * N_ + it * 16) * D_ + h * HD_ + nt * 16 + m16;
#pragma unroll
  for (int p = 0; p < 8; ++p) orow[(size_t)(p + 8 * g) * D_] = acc[p];
}

// ---------------------------------------------------------------------------
// Host launcher. Workspace layout (floats):
//   Q,K,V,A,B : 5 * 1024*256            = 1,310,720
//   S         : 2*4*512*512             = 2,097,152
//   O         : 1024*256                =   262,144
// Total ~14.0 MB.
// ---------------------------------------------------------------------------
extern "C" void kernel_launch(void* const* d_in, const int* in_sizes, int n_in,
                              void* d_out, int out_size, void* d_ws,
                              size_t ws_size, hipStream_t stream) {
  (void)in_sizes; (void)n_in; (void)out_size; (void)ws_size;
  const float* x   = (const float*)d_in[0];
  const int*   adj = (const int*)  d_in[1];
  const float* Wq  = (const float*)d_in[2];
  const float* bq  = (const float*)d_in[3];
  const float* Wk  = (const float*)d_in[4];
  const float* bk  = (const float*)d_in[5];
  const float* Wv  = (const float*)d_in[6];
  const float* bv  = (const float*)d_in[7];
  const float* Wo  = (const float*)d_in[8];
  const float* bo  = (const float*)d_in[9];
  const float* Wp1 = (const float*)d_in[10];  // [2D, D] row-major
  const float* bp1 = (const float*)d_in[11];
  const float* Wp2 = (const float*)d_in[12];  // [D, H] row-major
  const float* bp2 = (const float*)d_in[13];

  const int M = B_ * N_;  // 1024
  float* ws = (float*)d_ws;
  float* Qb = ws;
  float* Kb = Qb + (size_t)M * D_;
  float* Vb = Kb + (size_t)M * D_;
  float* Ab = Vb + (size_t)M * D_;
  float* Bb = Ab + (size_t)M * D_;
  float* Sb = Bb + (size_t)M * D_;                       // [B,H,N,N]
  float* Ob = Sb + (size_t)B_ * H_ * N_ * N_;

  const dim3 blk(256);
  const int gemm_blocks = (M / 16) * (D_ / 16) / 8;      // 128

  // Projections (bp1 folded into the x_i side of the physics MLP).
  gemm_xw_kernel<<<gemm_blocks, blk, 0, stream>>>(x, Wq, bq, Qb, M, D_, D_);
  gemm_xw_kernel<<<gemm_blocks, blk, 0, stream>>>(x, Wk, bk, Kb, M, D_, D_);
  gemm_xw_kernel<<<gemm_blocks, blk, 0, stream>>>(x, Wv, bv, Vb, M, D_, D_);
  gemm_xw_kernel<<<gemm_blocks, blk, 0, stream>>>(x, Wp1, bp1, Ab, M, D_, D_);
  gemm_xw_kernel<<<gemm_blocks, blk, 0, stream>>>(x, Wp1 + (size_t)D_ * D_,
                                                  nullptr, Bb, M, D_, D_);

  scores_kernel<<<B_ * 32 * 32, blk, 0, stream>>>(Qb, Kb, Ab, Bb, Wp2, bp2,
                                                  adj, Sb);
  softmax_kernel<<<(B_ * H_ * N_) / 8, blk, 0, stream>>>(Sb);
  pv_kernel<<<(B_ * H_ * 32 * 4) / 8, blk, 0, stream>>>(Sb, Vb, Ob);
  gemm_xw_kernel<<<gemm_blocks, blk, 0, stream>>>(Ob, Wo, bo, (float*)d_out,
                                                  M, D_, D_);
}